// SkipGramWordnetModel_50835232916094
// MI455X (gfx1250) — compile-verified
//
#include <hip/hip_runtime.h>
#include <math.h>

typedef float f32x2 __attribute__((ext_vector_type(2)));
typedef float f32x4 __attribute__((ext_vector_type(4)));
typedef float f32x8 __attribute__((ext_vector_type(8)));

#define DIM 128
#define WPB 8            // waves per block (wave32)
#define BLOCK (WPB * 32)

__device__ __forceinline__ float neg_logsig(float x) {
    // -log(sigmoid(x)) = softplus(-x), numerically stable
    float ax = fabsf(x);
    return fmaxf(-x, 0.0f) + log1pf(__expf(-ax));
}

// Compute 16 dot products <V[idx_i], u> (i = lane&15 supplies the gather index)
// via a chain of 32 V_WMMA_F32_16X16X4_F32 over K=128. B is the broadcast u row
// (all 16 columns identical), so D[i][j] == score_i for every j.
// Also accumulates this lane's half of |V[idx]|^2 into *cnorm_half.
__device__ __forceinline__ f32x8 tile_scores(const float* __restrict__ V,
                                             int idx,
                                             const float* s_urow,
                                             int lane,
                                             float* cnorm_half) {
    const float* cptr = V + (size_t)idx * DIM;
    const int off = 2 * (lane >> 4);   // lanes 0-15: K=0,1 ; lanes 16-31: K=2,3
    f32x8 acc = {};
    float nrm = 0.0f;
#pragma unroll 4
    for (int t = 0; t < 32; ++t) {
        f32x2 a = *(const f32x2*)(cptr + 4 * t + off);      // A fragment (gather)
        f32x2 b = *(const f32x2*)(s_urow + 4 * t + off);    // B fragment (u, LDS)
        nrm = fmaf(a.x, a.x, nrm);
        nrm = fmaf(a.y, a.y, nrm);
        acc = __builtin_amdgcn_wmma_f32_16x16x4_f32(
            /*neg_a=*/false, a, /*neg_b=*/false, b,
            /*c_mod=*/(short)0, acc, /*reuse_a=*/false, /*reuse_b=*/false);
    }
    *cnorm_half = nrm;
    // Result layout: lanes 0-15 hold scores for contexts 0..7 in acc[0..7],
    // lanes 16-31 hold contexts 8..15 (replicated across the 16 lanes of each half).
    return acc;
}

__global__ __launch_bounds__(BLOCK) void skipgram_wordnet_loss_kernel(
    const int* __restrict__ u, const int* __restrict__ v,
    const int* __restrict__ neg, const int* __restrict__ wn,
    const int* __restrict__ sim, const int* __restrict__ nsim,
    const int* __restrict__ mm, const float* __restrict__ U,
    const float* __restrict__ V, float* __restrict__ row_loss, int nrows) {
    __shared__ float s_u[WPB][DIM];

    const int lane = threadIdx.x & 31;
    const int wave = threadIdx.x >> 5;
    const int row = blockIdx.x * WPB + wave;
    const bool active = (row < nrows);

    float unorm2 = 0.0f;
    if (active) {
        int ui = u[row];
        const float* uptr = U + (size_t)ui * DIM;
        f32x4 uf = *(const f32x4*)(uptr + lane * 4);        // coalesced 512B/wave
        *(f32x4*)&s_u[wave][lane * 4] = uf;
        float p = fmaf(uf.x, uf.x, fmaf(uf.y, uf.y, fmaf(uf.z, uf.z, uf.w * uf.w)));
#pragma unroll
        for (int o = 16; o > 0; o >>= 1) p += __shfl_xor(p, o, 32);
        unorm2 = p;
    }
    __syncthreads();

    if (!active) return;

    const int slot = lane & 15;
    const int half_base = (lane >> 4) << 3;   // 0 or 8
    float tileAcc = 0.0f;                     // replicated x16 within each half

    // ---- w2v tiles: 48 padded slots: [0,8)=v(+), [8,28)=neg(-), [28,38)=wn(-), [38,48)=pad
#pragma unroll
    for (int g = 0; g < 3; ++g) {
        int s = 16 * g + slot;
        int idx = 0;
        if (s < 8)       idx = v[row * 8 + s];
        else if (s < 28) idx = neg[row * 20 + (s - 8)];
        else if (s < 38) idx = wn[row * 10 + (s - 28)];
        float nrm_unused;
        f32x8 sc = tile_scores(V, idx, s_u[wave], lane, &nrm_unused);
        int base = 16 * g + half_base;
#pragma unroll
        for (int r = 0; r < 8; ++r) {
            int sp = base + r;
            float sgn = (sp < 8) ? 1.0f : -1.0f;
            if (sp < 38) tileAcc += neg_logsig(sgn * sc[r]);
        }
    }

    // ---- distance tiles: sim (0.5*d^2), not_sim / mismatch (0.5*relu(1-d)^2)
#pragma unroll
    for (int q = 0; q < 3; ++q) {
        const int* arr = (q == 0) ? sim : ((q == 1) ? nsim : mm);
        int idx = arr[row * 16 + slot];
        float nrm;
        f32x8 sc = tile_scores(V, idx, s_u[wave], lane, &nrm);
        float cn = nrm + __shfl_xor(nrm, 16, 32);   // lanes i and i+16 hold |c_i|^2
#pragma unroll
        for (int r = 0; r < 8; ++r) {
            float cnr = __shfl(cn, half_base + r, 32);
            float d2 = fmaxf(unorm2 + cnr - 2.0f * sc[r], 0.0f);
            if (q == 0) {
                tileAcc += 0.5f * d2;
            } else {
                float h = fmaxf(1.0f - sqrtf(d2), 0.0f);
                tileAcc += 0.5f * h * h;
            }
        }
    }

    // lanes 0-15 share identical tileAcc (contexts 0-7 terms), lanes 16-31 likewise (8-15)
    float total = __shfl(tileAcc, 0, 32) + __shfl(tileAcc, 16, 32);
    if (lane == 0) row_loss[row] = total;
}

__global__ __launch_bounds__(256) void reduce_mean_kernel(
    const float* __restrict__ row_loss, float* __restrict__ out, int n) {
    __shared__ float sdata[256];
    float s = 0.0f;
    for (int i = threadIdx.x; i < n; i += 256) s += row_loss[i];
    sdata[threadIdx.x] = s;
    __syncthreads();
    for (int stride = 128; stride > 0; stride >>= 1) {
        if ((int)threadIdx.x < stride) sdata[threadIdx.x] += sdata[threadIdx.x + stride];
        __syncthreads();
    }
    if (threadIdx.x == 0) out[0] = sdata[0] / (float)n;
}

extern "C" void kernel_launch(void* const* d_in, const int* in_sizes, int n_in,
                              void* d_out, int out_size, void* d_ws, size_t ws_size,
                              hipStream_t stream) {
    const int*   u    = (const int*)d_in[0];
    const int*   v    = (const int*)d_in[1];
    const int*   neg  = (const int*)d_in[2];
    const int*   wn   = (const int*)d_in[3];
    const int*   sim  = (const int*)d_in[4];
    const int*   nsim = (const int*)d_in[5];
    const int*   mm   = (const int*)d_in[6];
    const float* U    = (const float*)d_in[7];
    const float* V    = (const float*)d_in[8];

    const int nrows = in_sizes[0];           // B = 4096
    float* row_loss = (float*)d_ws;          // nrows floats of scratch

    const int blocks = (nrows + WPB - 1) / WPB;
    skipgram_wordnet_loss_kernel<<<blocks, BLOCK, 0, stream>>>(
        u, v, neg, wn, sim, nsim, mm, U, V, row_loss, nrows);

    reduce_mean_kernel<<<1, 256, 0, stream>>>(row_loss, (float*)d_out, nrows);
}